// LinearTempNormLayer_4174708212416
// MI455X (gfx1250) — compile-verified
//
#include <hip/hip_runtime.h>
#include <math.h>

typedef __attribute__((ext_vector_type(2))) float v2f;
typedef __attribute__((ext_vector_type(8))) float v8f;

#define SEQ   2048
#define BATCH 32
#define HID   512
#define MTOT  (SEQ * BATCH)   /* 65536 rows of the GEMM */
#define KDIM  HID
#define NDIM  HID
#define EPSV  1e-4f

// ---------------------------------------------------------------------------
// GEMM: y[m][n] = sum_k x[m][k] * w[n][k] + bias[n]
// One wave computes a 32(M) x 32(N) tile with 2x2 V_WMMA_F32_16X16X4_F32
// accumulators. Block = 256 threads = 8 waves.
// ---------------------------------------------------------------------------
__global__ __launch_bounds__(256) void gemm_f32_wmma(
    const float* __restrict__ x,
    const float* __restrict__ w,
    const float* __restrict__ bias,
    float* __restrict__ y)
{
    const int lane      = threadIdx.x & 31;
    const int waveInBlk = threadIdx.x >> 5;
    const int waveId    = blockIdx.x * 8 + waveInBlk;

    const int nUnits = NDIM / 32;            // 16
    const int m0 = (waveId / nUnits) * 32;
    const int n0 = (waveId % nUnits) * 32;

    // A-matrix 16x4 f32 layout: lanes 0-15 -> M=lane, K pair {0,1};
    //                           lanes 16-31 -> M=lane-16, K pair {2,3}.
    // B-matrix 4x16 f32 layout: lane&15 -> N, lane half selects K row pair.
    const int mrow  = lane & 15;             // also the N index for B frags
    const int kLane = (lane >> 4) * 2;       // 0 or 2 within each K-chunk of 4

    const float* aPtr0 = x + (size_t)(m0 + mrow)      * KDIM + kLane;
    const float* aPtr1 = aPtr0 + (size_t)16 * KDIM;
    const float* bPtr0 = w + (size_t)(n0 + mrow)      * KDIM + kLane;
    const float* bPtr1 = bPtr0 + (size_t)16 * KDIM;

    v8f acc00 = {}; v8f acc01 = {}; v8f acc10 = {}; v8f acc11 = {};

#pragma unroll 4
    for (int kc = 0; kc < KDIM; kc += 4) {
        v2f a0 = *(const v2f*)(aPtr0 + kc);
        v2f a1 = *(const v2f*)(aPtr1 + kc);
        v2f b0 = *(const v2f*)(bPtr0 + kc);
        v2f b1 = *(const v2f*)(bPtr1 + kc);
        // 8 args: (neg_a, A, neg_b, B, c_mod, C, reuse_a, reuse_b)
        acc00 = __builtin_amdgcn_wmma_f32_16x16x4_f32(false, a0, false, b0,
                                                      (short)0, acc00, false, false);
        acc01 = __builtin_amdgcn_wmma_f32_16x16x4_f32(false, a0, false, b1,
                                                      (short)0, acc01, false, false);
        acc10 = __builtin_amdgcn_wmma_f32_16x16x4_f32(false, a1, false, b0,
                                                      (short)0, acc10, false, false);
        acc11 = __builtin_amdgcn_wmma_f32_16x16x4_f32(false, a1, false, b1,
                                                      (short)0, acc11, false, false);
    }

    // C/D 16x16 f32 layout: VGPR j -> row j (lanes 0-15) / row j+8 (lanes 16-31),
    // N = lane & 15 in both halves.
    const int nCol0 = n0 + (lane & 15);
    const int nCol1 = nCol0 + 16;
    const float bv0 = bias[nCol0];
    const float bv1 = bias[nCol1];
    const int mBase = m0 + ((lane >> 4) << 3);   // +8 for the upper lane half

#pragma unroll
    for (int j = 0; j < 8; ++j) {
        const size_t r0 = (size_t)(mBase + j)      * NDIM;
        const size_t r1 = (size_t)(mBase + j + 16) * NDIM;
        y[r0 + nCol0] = acc00[j] + bv0;
        y[r0 + nCol1] = acc01[j] + bv1;
        y[r1 + nCol0] = acc10[j] + bv0;
        y[r1 + nCol1] = acc11[j] + bv1;
    }
}

// ---------------------------------------------------------------------------
// Temporal-norm scan: one thread per (b,h) chain, in-place over y.
// Loop-carried dep is only mu/var (VALU); next timestep's load is
// software-pipelined into a register.
// ---------------------------------------------------------------------------
__global__ __launch_bounds__(256) void tempnorm_scan(
    const float* __restrict__ fgate,
    float* __restrict__ y,          // [SEQ][BATCH][HID], read + overwritten
    float* __restrict__ hidden)     // [BATCH][2*HID]
{
    const int t = blockIdx.x * blockDim.x + threadIdx.x;   // 0 .. B*H-1
    const int h = t & (HID - 1);
    const int b = t >> 9;                                  // t / HID

    const float f   = fgate[h];
    const float omf = 1.0f - f;

    float mu  = 0.0f;
    float var = 1.0f;

    float* p = y + (size_t)b * HID + h;
    const size_t stride = (size_t)BATCH * HID;

    float yv = *p;
    for (int s = 0; s < SEQ - 1; ++s) {
        const float ynext = p[stride];                 // prefetch next timestep
        const float diff  = yv - mu;
        const float outv  = tanhf(diff * rsqrtf(var + EPSV));
        *p  = outv;
        mu  = f * yv + omf * mu;
        var = f * (diff * diff) + omf * var;
        p  += stride;
        yv  = ynext;
    }
    {   // last timestep (no prefetch)
        const float diff = yv - mu;
        *p  = tanhf(diff * rsqrtf(var + EPSV));
        mu  = f * yv + omf * mu;
        var = f * (diff * diff) + omf * var;
    }

    hidden[(size_t)b * (2 * HID) + h]        = mu;
    hidden[(size_t)b * (2 * HID) + HID + h]  = var;
}

// ---------------------------------------------------------------------------
extern "C" void kernel_launch(void* const* d_in, const int* in_sizes, int n_in,
                              void* d_out, int out_size, void* d_ws, size_t ws_size,
                              hipStream_t stream)
{
    (void)in_sizes; (void)n_in; (void)out_size; (void)d_ws; (void)ws_size;

    const float* x     = (const float*)d_in[0];   // [SEQ,BATCH,HID]
    const float* w     = (const float*)d_in[1];   // [HID,HID]
    const float* bias  = (const float*)d_in[2];   // [HID]
    const float* fgate = (const float*)d_in[3];   // [HID]

    float* out    = (float*)d_out;
    float* y      = out;                               // y_out region, used in-place
    float* hidden = out + (size_t)SEQ * BATCH * HID;   // new_hidden region

    // GEMM: (MTOT/32)*(NDIM/32) = 32768 waves, 8 waves per block.
    const int totalWaves = (MTOT / 32) * (NDIM / 32);
    gemm_f32_wmma<<<dim3(totalWaves / 8), dim3(256), 0, stream>>>(x, w, bias, y);

    // Scan: one thread per (b,h) chain.
    tempnorm_scan<<<dim3((BATCH * HID) / 256), dim3(256), 0, stream>>>(fgate, y, hidden);
}